// GCN_81389630259984
// MI455X (gfx1250) — compile-verified
//
#include <hip/hip_runtime.h>
#include <math.h>

// GCN 2-layer forward for MI455X (gfx1250), wave32, WMMA f32 16x16x4.
#define IN_C  128
#define HID_C 64
#define OUT_C 16

typedef float v2f __attribute__((ext_vector_type(2)));
typedef float v8f __attribute__((ext_vector_type(8)));

__device__ __forceinline__ void atomAddF(float* p, float v) {
    // Lowers to global_atomic_add_f32 (no CAS loop) on gfx1250.
    unsafeAtomicAdd(p, v);
}

// ---------------- normalization ----------------
__global__ void k_init_deg(float* __restrict__ deg, int n) {
    int i = blockIdx.x * blockDim.x + threadIdx.x;
    if (i < n) deg[i] = 1.0f;  // self-loop weight folded in
}

__global__ void k_deg_scatter(const long long* __restrict__ ei,
                              const float* __restrict__ w,
                              float* __restrict__ deg, int e) {
    int i = blockIdx.x * blockDim.x + threadIdx.x;
    if (i < e) atomAddF(&deg[(int)ei[(long long)e + i]], w[i]);
}

__global__ void k_dinv(float* __restrict__ deg, int n) {
    int i = blockIdx.x * blockDim.x + threadIdx.x;
    if (i < n) {
        float d = deg[i];
        deg[i] = (d > 0.0f) ? rsqrtf(d) : 0.0f;
    }
}

__global__ void k_norm(const long long* __restrict__ ei,
                       const float* __restrict__ w,
                       const float* __restrict__ dinv,
                       float* __restrict__ norm, int e) {
    int i = blockIdx.x * blockDim.x + threadIdx.x;
    if (i < e) {
        int r = (int)ei[i];
        int c = (int)ei[(long long)e + i];
        norm[i] = dinv[r] * w[i] * dinv[c];
    }
}

// ---------------- GEMM1: H1 = X @ W1, AGG1 = H1 * dinv^2 ----------------
// LDS holds W1 pre-swizzled into k-pairs: P[k2][n] = {W[2k2][n], W[2k2+1][n]}
// so each B fragment is one contiguous ds_load_b64.
__global__ __launch_bounds__(256) void k_gemm1(const float* __restrict__ X,
                                               const float* __restrict__ W,
                                               const float* __restrict__ dinv,
                                               float* __restrict__ H,
                                               float* __restrict__ AGG, int ntiles) {
    __shared__ v2f wp[(IN_C / 2) * HID_C];  // 4096 pairs = 32 KB
    const int t = threadIdx.x;
#pragma unroll
    for (int j = 0; j < 16; ++j) {
        int i = t + j * 256;               // pair index
        int k2 = i >> 6;                   // k/2
        int n  = i & 63;
        v2f p;
        p.x = W[(k2 * 2) * HID_C + n];
        p.y = W[(k2 * 2 + 1) * HID_C + n];
        wp[i] = p;
    }
    __syncthreads();

    const int wave = t >> 5, lane = t & 31;
    const int tile = blockIdx.x * 8 + wave;
    if (tile >= ntiles) return;

    const int m     = lane & 15;           // A row / B col within tile
    const int khalf = lane >> 4;           // 0 or 1 -> K offset 0 or 2
    const float* xr = X + (size_t)(tile * 16 + m) * IN_C + (khalf << 1);

    v8f c0 = {}, c1 = {}, c2 = {}, c3 = {};
#pragma unroll 8
    for (int k = 0; k < IN_C; k += 4) {
        v2f a = *(const v2f*)(xr + k);     // contiguous global_load_b64
        const v2f* bp = wp + ((k >> 1) + khalf) * HID_C + m;
        v2f b0 = bp[0];
        v2f b1 = bp[16];
        v2f b2 = bp[32];
        v2f b3 = bp[48];
        c0 = __builtin_amdgcn_wmma_f32_16x16x4_f32(false, a, false, b0, (short)0, c0, false, false);
        c1 = __builtin_amdgcn_wmma_f32_16x16x4_f32(false, a, false, b1, (short)0, c1, false, false);
        c2 = __builtin_amdgcn_wmma_f32_16x16x4_f32(false, a, false, b2, (short)0, c2, false, false);
        c3 = __builtin_amdgcn_wmma_f32_16x16x4_f32(false, a, false, b3, (short)0, c3, false, false);
    }

    // Epilogue: write H1 and self-loop contribution AGG1 = H1*dinv^2.
    const int rbase = khalf << 3;          // 0 or 8
#pragma unroll
    for (int r = 0; r < 8; ++r) {
        const int row = tile * 16 + rbase + r;
        float d = dinv[row];
        float s = d * d;
        float* oh = H   + (size_t)row * HID_C + m;
        float* oa = AGG + (size_t)row * HID_C + m;
        oh[0]  = c0[r];      oa[0]  = c0[r] * s;
        oh[16] = c1[r];      oa[16] = c1[r] * s;
        oh[32] = c2[r];      oa[32] = c2[r] * s;
        oh[48] = c3[r];      oa[48] = c3[r] * s;
    }
}

// ---------------- layer-1 edge scatter ----------------
__global__ void k_scat1(const long long* __restrict__ ei, const float* __restrict__ norm,
                        const float* __restrict__ H, float* __restrict__ agg, int e) {
    long long i = (long long)blockIdx.x * blockDim.x + threadIdx.x;  // e*16 threads
    int ed = (int)(i >> 4);
    if (ed >= e) return;
    int c4 = ((int)i & 15) << 2;
    int r = (int)ei[ed];
    int c = (int)ei[(long long)e + ed];
    float s = norm[ed];
    float4 h = *(const float4*)(H + (size_t)r * HID_C + c4);
    float* o = agg + (size_t)c * HID_C + c4;
    atomAddF(o + 0, h.x * s);
    atomAddF(o + 1, h.y * s);
    atomAddF(o + 2, h.z * s);
    atomAddF(o + 3, h.w * s);
}

// ---------------- GEMM2: H2 = relu(AGG1) @ W2, AGG2 = H2 * dinv^2 ----------------
__global__ __launch_bounds__(256) void k_gemm2(const float* __restrict__ A,
                                               const float* __restrict__ W,
                                               const float* __restrict__ dinv,
                                               float* __restrict__ H,
                                               float* __restrict__ AGG, int ntiles) {
    __shared__ v2f wp[(HID_C / 2) * OUT_C];  // 512 pairs = 4 KB
    const int t = threadIdx.x;
    if (t < (HID_C / 2) * OUT_C) {
        int k2 = t >> 4;
        int n  = t & 15;
        v2f p;
        p.x = W[(k2 * 2) * OUT_C + n];
        p.y = W[(k2 * 2 + 1) * OUT_C + n];
        wp[t] = p;
    }
    __syncthreads();

    const int wave = t >> 5, lane = t & 31;
    const int tile = blockIdx.x * 8 + wave;
    if (tile >= ntiles) return;

    const int m     = lane & 15;
    const int khalf = lane >> 4;
    const float* ar = A + (size_t)(tile * 16 + m) * HID_C + (khalf << 1);

    v8f c = {};
#pragma unroll 8
    for (int k = 0; k < HID_C; k += 4) {
        v2f a;
        a.x = fmaxf(ar[k], 0.0f);           // fused ReLU
        a.y = fmaxf(ar[k + 1], 0.0f);
        v2f b = wp[((k >> 1) + khalf) * OUT_C + m];  // one ds_load_b64
        c = __builtin_amdgcn_wmma_f32_16x16x4_f32(false, a, false, b, (short)0, c, false, false);
    }

    const int rbase = khalf << 3;
#pragma unroll
    for (int r = 0; r < 8; ++r) {
        const int row = tile * 16 + rbase + r;
        float d = dinv[row];
        float s = d * d;
        H  [(size_t)row * OUT_C + m] = c[r];
        AGG[(size_t)row * OUT_C + m] = c[r] * s;
    }
}

// ---------------- layer-2 edge scatter ----------------
__global__ void k_scat2(const long long* __restrict__ ei, const float* __restrict__ norm,
                        const float* __restrict__ H, float* __restrict__ agg, int e) {
    long long i = (long long)blockIdx.x * blockDim.x + threadIdx.x;  // e*4 threads
    int ed = (int)(i >> 2);
    if (ed >= e) return;
    int c4 = ((int)i & 3) << 2;
    int r = (int)ei[ed];
    int c = (int)ei[(long long)e + ed];
    float s = norm[ed];
    float4 h = *(const float4*)(H + (size_t)r * OUT_C + c4);
    float* o = agg + (size_t)c * OUT_C + c4;
    atomAddF(o + 0, h.x * s);
    atomAddF(o + 1, h.y * s);
    atomAddF(o + 2, h.z * s);
    atomAddF(o + 3, h.w * s);
}

// ---------------- row-wise log_softmax over 16 channels ----------------
__global__ void k_lsm(const float* __restrict__ agg, float* __restrict__ out, int n) {
    int i = blockIdx.x * blockDim.x + threadIdx.x;
    if (i >= n) return;
    const float4* a = (const float4*)(agg + (size_t)i * OUT_C);
    float4 v0 = a[0], v1 = a[1], v2 = a[2], v3 = a[3];
    float mx = v0.x;
    mx = fmaxf(mx, v0.y); mx = fmaxf(mx, v0.z); mx = fmaxf(mx, v0.w);
    mx = fmaxf(mx, v1.x); mx = fmaxf(mx, v1.y); mx = fmaxf(mx, v1.z); mx = fmaxf(mx, v1.w);
    mx = fmaxf(mx, v2.x); mx = fmaxf(mx, v2.y); mx = fmaxf(mx, v2.z); mx = fmaxf(mx, v2.w);
    mx = fmaxf(mx, v3.x); mx = fmaxf(mx, v3.y); mx = fmaxf(mx, v3.z); mx = fmaxf(mx, v3.w);
    float s = 0.0f;
    s += expf(v0.x - mx) + expf(v0.y - mx) + expf(v0.z - mx) + expf(v0.w - mx);
    s += expf(v1.x - mx) + expf(v1.y - mx) + expf(v1.z - mx) + expf(v1.w - mx);
    s += expf(v2.x - mx) + expf(v2.y - mx) + expf(v2.z - mx) + expf(v2.w - mx);
    s += expf(v3.x - mx) + expf(v3.y - mx) + expf(v3.z - mx) + expf(v3.w - mx);
    float lse = mx + logf(s);
    float4* o = (float4*)(out + (size_t)i * OUT_C);
    float4 r0, r1, r2, r3;
    r0.x = v0.x - lse; r0.y = v0.y - lse; r0.z = v0.z - lse; r0.w = v0.w - lse;
    r1.x = v1.x - lse; r1.y = v1.y - lse; r1.z = v1.z - lse; r1.w = v1.w - lse;
    r2.x = v2.x - lse; r2.y = v2.y - lse; r2.z = v2.z - lse; r2.w = v2.w - lse;
    r3.x = v3.x - lse; r3.y = v3.y - lse; r3.z = v3.z - lse; r3.w = v3.w - lse;
    o[0] = r0; o[1] = r1; o[2] = r2; o[3] = r3;
}

extern "C" void kernel_launch(void* const* d_in, const int* in_sizes, int n_in,
                              void* d_out, int out_size, void* d_ws, size_t ws_size,
                              hipStream_t stream) {
    const float*     x  = (const float*)d_in[0];
    const long long* ei = (const long long*)d_in[1];   // int64 [2,E]
    const float*     ew = (const float*)d_in[2];
    const float*     W1 = (const float*)d_in[3];
    const float*     W2 = (const float*)d_in[4];
    float* out = (float*)d_out;

    const int n = in_sizes[0] / IN_C;   // 100000
    const int e = in_sizes[2];          // 1600000

    // workspace layout (all float, 16B-aligned slices)
    float* deg  = (float*)d_ws;                 // n   (becomes dinv in place)
    float* nrm  = deg + n;                      // e
    float* H1   = nrm + e;                      // n*64
    float* AGG1 = H1 + (size_t)n * HID_C;       // n*64
    float* H2   = AGG1 + (size_t)n * HID_C;     // n*16
    float* AGG2 = H2 + (size_t)n * OUT_C;       // n*16

    const int ntiles = n / 16;                  // n is a multiple of 16
    const int B = 256;

    k_init_deg   <<<(n + B - 1) / B, B, 0, stream>>>(deg, n);
    k_deg_scatter<<<(e + B - 1) / B, B, 0, stream>>>(ei, ew, deg, e);
    k_dinv       <<<(n + B - 1) / B, B, 0, stream>>>(deg, n);
    k_norm       <<<(e + B - 1) / B, B, 0, stream>>>(ei, ew, deg, nrm, e);

    // GEMM1 fuses the self-loop init of AGG1; scatter adds edge messages.
    k_gemm1      <<<(ntiles + 7) / 8, B, 0, stream>>>(x, W1, deg, H1, AGG1, ntiles);
    k_scat1      <<<((size_t)e * 16 + B - 1) / B, B, 0, stream>>>(ei, nrm, H1, AGG1, e);

    // GEMM2 fuses ReLU on load and the self-loop init of AGG2.
    k_gemm2      <<<(ntiles + 7) / 8, B, 0, stream>>>(AGG1, W2, deg, H2, AGG2, ntiles);
    k_scat2      <<<((size_t)e * 4 + B - 1) / B, B, 0, stream>>>(ei, nrm, H2, AGG2, e);

    k_lsm        <<<(n + B - 1) / B, B, 0, stream>>>(AGG2, out, n);
}